// AudioS4_18262200942824
// MI455X (gfx1250) — compile-verified
//
#include <hip/hip_runtime.h>
#include <hip/hip_bf16.h>
#include <math.h>

// ---------------- model constants ----------------
#define B_  8
#define L_  8192
#define D_  64
#define H_  128
#define N_  64        // SSM state size
#define T_  64        // chunk length
#define CN_ 128       // chunks = L_/T_
#define G_  256       // 2*H

typedef __attribute__((ext_vector_type(2))) float v2f;
typedef __attribute__((ext_vector_type(8))) float v8f;

// CDNA5 f32 tensor-core op: D(16x16) = A(16x4) * B(4x16) + C
__device__ __forceinline__ v8f wmma4(v2f a, v2f b, v8f c) {
  return __builtin_amdgcn_wmma_f32_16x16x4_f32(false, a, false, b, (short)0, c,
                                               false, false);
}

// ======================================================================
// Generic WMMA GEMM with bias: C(MxN) = A(MxK)@B(KxN) + bias(N)
// one wave per 16x16 output tile; K multiple of 4
// ======================================================================
__global__ void gemm_bias_kernel(const float* __restrict__ A,
                                 const float* __restrict__ Bm,
                                 const float* __restrict__ bias,
                                 float* __restrict__ C,
                                 int M, int K, int N) {
  int lane = threadIdx.x & 31;
  int n0 = blockIdx.x * 16;
  int m0 = blockIdx.y * 16;
  int lm = lane & 15;
  int kh = (lane >> 4) * 2;          // lanes 16-31 hold K+2,K+3
  v8f acc = {};
  for (int k = 0; k < K; k += 4) {
    int kk = k + kh;
    v2f af, bf;
    af.x = A[(size_t)(m0 + lm) * K + kk];
    af.y = A[(size_t)(m0 + lm) * K + kk + 1];
    bf.x = Bm[(size_t)kk * N + n0 + lm];
    bf.y = Bm[(size_t)(kk + 1) * N + n0 + lm];
    acc = wmma4(af, bf, acc);
  }
  int col = n0 + lm;
  float bv = bias[col];
  int rbase = m0 + ((lane >> 4) << 3);   // lanes>=16 hold rows M+8..M+15
#pragma unroll
  for (int r = 0; r < 8; ++r)
    C[(size_t)(rbase + r) * N + col] = acc[r] + bv;
}

// ======================================================================
// Per-layer SSM parameter precompute:
//  w = exp(dt*A), wT = w^64, C' = (Cre+iCim)*expm1(dtA)/A
//  A_h (64 x 192):  [l][j]      = Toeplitz  2Re(Sum C' w^(l-j))
//                   [l][64+n]   =  2Re(C' w^(l+1))
//                   [l][128+n]  = -2Im(C' w^(l+1))
// ======================================================================
__global__ void precompute_kernel(const float* __restrict__ log_dt,
                                  const float* __restrict__ log_A_real,
                                  const float* __restrict__ A_imag,
                                  const float* __restrict__ C_re,
                                  const float* __restrict__ C_im,
                                  float* __restrict__ Wbuf,   // (H,N,2)
                                  float* __restrict__ WTbuf,  // (H,N,2)
                                  float* __restrict__ Abuf) { // (H,64,192)
  int h = blockIdx.x;
  int n = threadIdx.x;               // 64 threads
  __shared__ float red[N_];
  __shared__ float Ks[T_];
  int idx = h * N_ + n;
  float dt = __expf(log_dt[h]);
  float Ar = -__expf(log_A_real[idx]);
  float Ai = A_imag[idx];
  float dr = dt * Ar, di = dt * Ai;
  float er = __expf(dr);
  float wr = er * __cosf(di), wi = er * __sinf(di);
  float emr = wr - 1.0f, emi = wi;          // expm1(dtA)
  float cr = C_re[idx], ci = C_im[idx];
  float tr = cr * emr - ci * emi;
  float ti = cr * emi + ci * emr;
  float den = Ar * Ar + Ai * Ai;
  float cpr = (tr * Ar + ti * Ai) / den;    // C' = t * conj(A)/|A|^2
  float cpi = (ti * Ar - tr * Ai) / den;
  Wbuf[idx * 2] = wr; Wbuf[idx * 2 + 1] = wi;
  float sr = wr, si = wi;                   // wT = w^64 via 6 squarings
#pragma unroll
  for (int t = 0; t < 6; ++t) { float nr = sr*sr - si*si; si = 2.0f*sr*si; sr = nr; }
  WTbuf[idx * 2] = sr; WTbuf[idx * 2 + 1] = si;

  float pr = 1.0f, pi = 0.0f;               // w^l
  float* Ah = Abuf + (size_t)h * T_ * 192;
  for (int l = 0; l < T_; ++l) {
    red[n] = cpr * pr - cpi * pi;
    __syncthreads();
    if (n == 0) { float s = 0.f; for (int j = 0; j < N_; ++j) s += red[j]; Ks[l] = 2.0f * s; }
    float nr = pr * wr - pi * wi;           // advance to w^(l+1)
    float ni = pr * wi + pi * wr;
    pr = nr; pi = ni;
    Ah[l * 192 + 64 + n]  =  2.0f * (cpr * pr - cpi * pi);
    Ah[l * 192 + 128 + n] = -2.0f * (cpr * pi + cpi * pr);
    __syncthreads();
  }
  __syncthreads();
  for (int t = n; t < T_ * T_; t += N_) {   // lower-triangular Toeplitz
    int r = t >> 6, c = t & 63;
    Ah[r * 192 + c] = (c <= r) ? Ks[r - c] : 0.0f;
  }
}

// ======================================================================
// (B,L,H) -> (B,H,L) tiled transpose
// ======================================================================
__global__ void transpose_kernel(const float* __restrict__ in,
                                 float* __restrict__ out) {
  __shared__ float tile[32][33];
  int b = blockIdx.z;
  int l0 = blockIdx.x * 32;
  int h0 = blockIdx.y * 32;
  int tx = threadIdx.x, ty = threadIdx.y;
  for (int i = ty; i < 32; i += 8)
    tile[i][tx] = in[(size_t)b * L_ * H_ + (size_t)(l0 + i) * H_ + h0 + tx];
  __syncthreads();
  for (int i = ty; i < 32; i += 8)
    out[(size_t)b * H_ * L_ + (size_t)(h0 + i) * L_ + l0 + tx] = tile[tx][i];
}

// ======================================================================
// chunk summaries: q[b,h,c,n] = sum_j w^(T-1-j) u[cT+j]  (Horner)
// one wave per (b,h,c); lane owns states n=lane and n=lane+32
// ======================================================================
__global__ void chunk_summary_kernel(const float* __restrict__ U,
                                     const float* __restrict__ Wbuf,
                                     float* __restrict__ Q) {
  int gwave = (int)((blockIdx.x * blockDim.x + threadIdx.x) >> 5);
  int lane = threadIdx.x & 31;
  int c = gwave & (CN_ - 1);
  int bh = gwave >> 7;                 // b*H + h
  int hh = bh & (H_ - 1);
  const float* u = U + (size_t)bh * L_ + c * T_;
  float w0r = Wbuf[(hh * N_ + lane) * 2],      w0i = Wbuf[(hh * N_ + lane) * 2 + 1];
  float w1r = Wbuf[(hh * N_ + lane + 32) * 2], w1i = Wbuf[(hh * N_ + lane + 32) * 2 + 1];
  float q0r = 0.f, q0i = 0.f, q1r = 0.f, q1i = 0.f;
  for (int j = 0; j < T_; ++j) {
    float uv = u[j];
    float a = q0r * w0r - q0i * w0i + uv;
    float bimag = q0r * w0i + q0i * w0r;
    q0r = a; q0i = bimag;
    a     = q1r * w1r - q1i * w1i + uv;
    bimag = q1r * w1i + q1i * w1r;
    q1r = a; q1i = bimag;
  }
  float* q = Q + (size_t)gwave * 128;  // [0..63]=re, [64..127]=im
  q[lane] = q0r; q[lane + 32] = q1r;
  q[64 + lane] = q0i; q[96 + lane] = q1i;
}

// ======================================================================
// chunk prefix scan: s_pre[c] = wT * s_pre[c-1] + q[c-1], s_pre[0]=0
// one wave per (b,h)
// ======================================================================
__global__ void chunk_scan_kernel(const float* __restrict__ Q,
                                  const float* __restrict__ WTbuf,
                                  float* __restrict__ SP) {
  int gwave = (int)((blockIdx.x * blockDim.x + threadIdx.x) >> 5);  // b*H+h
  int lane = threadIdx.x & 31;
  int hh = gwave & (H_ - 1);
  float w0r = WTbuf[(hh * N_ + lane) * 2],      w0i = WTbuf[(hh * N_ + lane) * 2 + 1];
  float w1r = WTbuf[(hh * N_ + lane + 32) * 2], w1i = WTbuf[(hh * N_ + lane + 32) * 2 + 1];
  float s0r = 0.f, s0i = 0.f, s1r = 0.f, s1i = 0.f;
  const float* q = Q + (size_t)gwave * CN_ * 128;
  float* sp = SP + (size_t)gwave * CN_ * 128;
  for (int c = 0; c < CN_; ++c) {
    float* d = sp + c * 128;
    d[lane] = s0r; d[lane + 32] = s1r; d[64 + lane] = s0i; d[96 + lane] = s1i;
    const float* qc = q + c * 128;
    float q0r = qc[lane], q1r = qc[lane + 32], q0i = qc[64 + lane], q1i = qc[96 + lane];
    float a = s0r * w0r - s0i * w0i + q0r;
    float bimag = s0r * w0i + s0i * w0r + q0i;
    s0r = a; s0i = bimag;
    a     = s1r * w1r - s1i * w1i + q1r;
    bimag = s1r * w1i + s1i * w1r + q1i;
    s1r = a; s1i = bimag;
  }
}

__device__ __forceinline__ float gelu_tanh(float x) {
  float t = tanhf(0.7978845608028654f * (x + 0.044715f * x * x * x));
  return 0.5f * x * (1.0f + t);
}

// ======================================================================
// conv GEMM per h: Y(64x1024) = A_h(64x192) @ [u_chunk; s_re; s_im]
// + D_skip*u, gelu.  cols = b*128 + c.  4 waves/block = 4 m-tiles.
// ======================================================================
__global__ void conv_gemm_kernel(const float* __restrict__ Abuf,
                                 const float* __restrict__ U,
                                 const float* __restrict__ SP,
                                 const float* __restrict__ Dskip,
                                 float* __restrict__ Y) {
  int wave = threadIdx.x >> 5;
  int lane = threadIdx.x & 31;
  int h = blockIdx.y;
  int n0 = blockIdx.x * 16;
  int m0 = wave * 16;
  int lm = lane & 15;
  int kh = (lane >> 4) * 2;
  const float* Ah = Abuf + (size_t)h * T_ * 192;
  int col = n0 + lm;                   // b*128 + c
  int b = col >> 7, c = col & 127;
  const float* ucol = U + (size_t)(b * H_ + h) * L_ + c * T_;
  const float* spcol = SP + ((size_t)(b * H_ + h) * CN_ + c) * 128;
  v8f acc = {};
  for (int k = 0; k < 192; k += 4) {
    int kk = k + kh;
    v2f af, bf;
    af.x = Ah[(m0 + lm) * 192 + kk];
    af.y = Ah[(m0 + lm) * 192 + kk + 1];
    bf.x = (kk < 64) ? ucol[kk] : spcol[kk - 64];
    bf.y = (kk + 1 < 64) ? ucol[kk + 1] : spcol[kk - 63];
    acc = wmma4(af, bf, acc);
  }
  float dsk = Dskip[h];
  int rbase = m0 + ((lane >> 4) << 3);
#pragma unroll
  for (int r = 0; r < 8; ++r) {
    int l = c * T_ + rbase + r;
    float uv = U[(size_t)(b * H_ + h) * L_ + l];
    float x = acc[r] + dsk * uv;
    Y[(size_t)(b * H_ + h) * L_ + l] = gelu_tanh(x);
  }
}

// ======================================================================
// GLU: per b, a = Wg[0:128]@Y_b + ba ; g = Wg[128:256]@Y_b + bg
// P = a * sigmoid(g).  8 waves/block cover all 8 m-tiles of H.
// ======================================================================
__global__ void glu_gemm_kernel(const float* __restrict__ Wg,
                                const float* __restrict__ bg,
                                const float* __restrict__ Y,
                                float* __restrict__ P) {
  int wave = threadIdx.x >> 5;
  int lane = threadIdx.x & 31;
  int b = blockIdx.y;
  int n0 = blockIdx.x * 16;            // along L
  int m0 = wave * 16;                  // output h
  int lm = lane & 15;
  int kh = (lane >> 4) * 2;
  const float* Yb = Y + (size_t)b * H_ * L_;
  v8f accA = {}, accG = {};
  for (int k = 0; k < H_; k += 4) {
    int kk = k + kh;
    v2f a1, a2, bf;
    a1.x = Wg[(m0 + lm) * H_ + kk];
    a1.y = Wg[(m0 + lm) * H_ + kk + 1];
    a2.x = Wg[(m0 + lm + H_) * H_ + kk];
    a2.y = Wg[(m0 + lm + H_) * H_ + kk + 1];
    bf.x = Yb[(size_t)kk * L_ + n0 + lm];
    bf.y = Yb[(size_t)(kk + 1) * L_ + n0 + lm];
    accA = wmma4(a1, bf, accA);
    accG = wmma4(a2, bf, accG);
  }
  int colg = n0 + lm;
  int rbase = m0 + ((lane >> 4) << 3);
#pragma unroll
  for (int r = 0; r < 8; ++r) {
    int hrow = rbase + r;
    float av = accA[r] + bg[hrow];
    float gv = accG[r] + bg[hrow + H_];
    float s = 1.0f / (1.0f + __expf(-gv));
    P[(size_t)(b * H_ + hrow) * L_ + colg] = av * s;
  }
}

// ======================================================================
// residual + RMSNorm: h[b,l,:] = normalize(P[b,:,l] + h[b,l,:]) * sqrt(H)
// one block (128 threads) per (b,l)
// ======================================================================
__global__ void resid_rmsnorm_kernel(const float* __restrict__ P,
                                     float* __restrict__ Hio) {
  int bl = blockIdx.x;                 // b*L + l
  int b = bl >> 13, l = bl & (L_ - 1);
  int h = threadIdx.x;
  __shared__ float red[4];
  float v = P[(size_t)(b * H_ + h) * L_ + l] + Hio[(size_t)bl * H_ + h];
  float ss = v * v;
#pragma unroll
  for (int off = 16; off > 0; off >>= 1) ss += __shfl_xor(ss, off, 32);
  if ((threadIdx.x & 31) == 0) red[threadIdx.x >> 5] = ss;
  __syncthreads();
  float tot = red[0] + red[1] + red[2] + red[3];
  float nrm = sqrtf(tot);
  float scale = 11.313708498984761f / fmaxf(nrm, 1e-12f);   // sqrt(128)
  Hio[(size_t)bl * H_ + h] = v * scale;
}

// ======================================================================
extern "C" void kernel_launch(void* const* d_in, const int* in_sizes, int n_in,
                              void* d_out, int out_size, void* d_ws, size_t ws_size,
                              hipStream_t stream) {
  const float* x        = (const float*)d_in[0];
  const float* W_in     = (const float*)d_in[1];
  const float* b_in     = (const float*)d_in[2];
  const float* log_dt   = (const float*)d_in[3];
  const float* log_Ar   = (const float*)d_in[4];
  const float* A_imag   = (const float*)d_in[5];
  const float* C_re     = (const float*)d_in[6];
  const float* C_im     = (const float*)d_in[7];
  const float* D_skip   = (const float*)d_in[8];
  const float* W_glu    = (const float*)d_in[9];
  const float* b_glu    = (const float*)d_in[10];
  const float* W_out    = (const float*)d_in[11];
  const float* b_out    = (const float*)d_in[12];
  float* out = (float*)d_out;

  // workspace carve-up (floats)
  float* ws = (float*)d_ws;
  size_t o = 0;
  float* hbuf  = ws + o; o += (size_t)B_ * L_ * H_;          // 8.4M
  float* ubuf  = ws + o; o += (size_t)B_ * L_ * H_;          // 8.4M
  float* ybuf  = ws + o; o += (size_t)B_ * L_ * H_;          // 8.4M
  float* qbuf  = ws + o; o += (size_t)B_ * H_ * CN_ * 128;   // 16.8M
  float* spbuf = ws + o; o += (size_t)B_ * H_ * CN_ * 128;   // 16.8M
  float* abuf  = ws + o; o += (size_t)H_ * T_ * 192;         // 1.6M
  float* wbuf  = ws + o; o += (size_t)H_ * N_ * 2;
  float* wtbuf = ws + o; o += (size_t)H_ * N_ * 2;

  const int M = B_ * L_;   // 65536

  // input projection: h = x @ W_in + b_in   (M x 64) @ (64 x 128)
  gemm_bias_kernel<<<dim3(H_ / 16, M / 16), 32, 0, stream>>>(
      x, W_in, b_in, hbuf, M, D_, H_);

  for (int i = 0; i < 3; ++i) {
    precompute_kernel<<<H_, N_, 0, stream>>>(
        log_dt + (size_t)i * H_, log_Ar + (size_t)i * H_ * N_,
        A_imag + (size_t)i * H_ * N_, C_re + (size_t)i * H_ * N_,
        C_im + (size_t)i * H_ * N_, wbuf, wtbuf, abuf);

    transpose_kernel<<<dim3(L_ / 32, H_ / 32, B_), dim3(32, 8), 0, stream>>>(
        hbuf, ubuf);

    chunk_summary_kernel<<<(B_ * H_ * CN_) / 8, 256, 0, stream>>>(
        ubuf, wbuf, qbuf);

    chunk_scan_kernel<<<(B_ * H_) / 8, 256, 0, stream>>>(qbuf, wtbuf, spbuf);

    conv_gemm_kernel<<<dim3(64, H_), 128, 0, stream>>>(
        abuf, ubuf, spbuf, D_skip + (size_t)i * H_, ybuf);

    glu_gemm_kernel<<<dim3(L_ / 16, B_), 256, 0, stream>>>(
        W_glu + (size_t)i * G_ * H_, b_glu + (size_t)i * G_, ybuf, ubuf);

    resid_rmsnorm_kernel<<<B_ * L_, H_, 0, stream>>>(ubuf, hbuf);
  }

  // output projection: out = h @ W_out + b_out  (M x 128) @ (128 x 64)
  gemm_bias_kernel<<<dim3(D_ / 16, M / 16), 32, 0, stream>>>(
      hbuf, W_out, b_out, out, M, H_, D_);
}